// GCNConvLayer_78847009620726
// MI455X (gfx1250) — compile-verified
//
#include <hip/hip_runtime.h>

// Problem constants (also derived from in_sizes at launch for robustness)
#define D_IN  128
#define D_OUT 64

typedef __attribute__((ext_vector_type(2))) float v2f;
typedef __attribute__((ext_vector_type(8))) float v8f;

// ---------------------------------------------------------------------------
// 1) deg[n] = 1.0 (self loop)
__global__ void deg_init_kernel(float* __restrict__ deg, int n) {
    int i = blockIdx.x * blockDim.x + threadIdx.x;
    if (i < n) deg[i] = 1.0f;
}

// 2) deg[to[e]] += 1 per edge
__global__ void deg_count_kernel(const int* __restrict__ to_idx,
                                 float* __restrict__ deg, int e) {
    int i = blockIdx.x * blockDim.x + threadIdx.x;
    if (i < e) atomicAdd(&deg[to_idx[i]], 1.0f);
}

// 3) dinv[n] = rsqrt(deg[n]) in place (deg >= 1 always, keep guard anyway)
__global__ void dinv_kernel(float* __restrict__ deg, int n) {
    int i = blockIdx.x * blockDim.x + threadIdx.x;
    if (i < n) {
        float d = deg[i];
        deg[i] = (d > 0.0f) ? rsqrtf(d) : 0.0f;
    }
}

// ---------------------------------------------------------------------------
// 4) h = x @ W^T + b via V_WMMA_F32_16X16X4_F32 (fp32 matrix pipe).
//    Block = 128 threads = 4 waves. Each wave owns one 16x16 output tile:
//    rows = 16 nodes (blockIdx.x), cols = 16 outputs (wave id * 16).
//    A (16x4 f32): lane<16 -> (M=lane, K=k..k+1); lane>=16 -> (M=lane-16, K=k+2..k+3)
//    B (4x16 f32): mirrors A with B[K][N] = W[n0+N][K]
//    C/D (16x16):  VGPR r, lane L -> (M = r + 8*(L>>4), N = L&15)
__global__ void __launch_bounds__(128)
gemm_wmma_kernel(const float* __restrict__ x, const float* __restrict__ W,
                 const float* __restrict__ b, float* __restrict__ h, int n_nodes) {
    const int lane = threadIdx.x & 31;
    const int wave = threadIdx.x >> 5;       // 0..3 -> N tile
    const int m0   = blockIdx.x * 16;        // node tile base
    const int n0   = wave * 16;              // output-channel tile base
    const int half = lane >> 4;              // 0 or 1
    const int l    = lane & 15;

    // Clamp row reads for a ragged tail tile (stores are guarded below).
    int row = m0 + l;
    int rrow = (row < n_nodes) ? row : (n_nodes - 1);

    const float* __restrict__ xrow = x + (size_t)rrow * D_IN;  // A source row
    const float* __restrict__ wrow = W + (size_t)(n0 + l) * D_IN; // B source row

    // Accumulator pre-loaded with bias: element (M, N=l) for every M.
    float bias = b[n0 + l];
    v8f acc;
#pragma unroll
    for (int r = 0; r < 8; ++r) acc[r] = bias;

#pragma unroll
    for (int k = 0; k < D_IN; k += 4) {
        v2f a, bb;
        // contiguous float2 per lane
        a[0]  = xrow[k + 2 * half + 0];
        a[1]  = xrow[k + 2 * half + 1];
        bb[0] = wrow[k + 2 * half + 0];
        bb[1] = wrow[k + 2 * half + 1];
        acc = __builtin_amdgcn_wmma_f32_16x16x4_f32(
            /*neg_a=*/false, a, /*neg_b=*/false, bb,
            /*c_mod=*/(short)0, acc, /*reuse_a=*/false, /*reuse_b=*/false);
    }

    // Store D: row m0 + r + 8*half, col n0 + l
    float* __restrict__ hbase = h + (size_t)m0 * D_OUT + n0;
#pragma unroll
    for (int r = 0; r < 8; ++r) {
        int mr = r + 8 * half;
        if (m0 + mr < n_nodes)
            hbase[(size_t)mr * D_OUT + l] = acc[r];
    }
}

// ---------------------------------------------------------------------------
// 5) out[n][d] = h[n][d] * dinv[n]^2   (self-loop term; initializes d_out)
__global__ void selfloop_kernel(const float* __restrict__ h,
                                const float* __restrict__ dinv,
                                float* __restrict__ out, int n) {
    int t = blockIdx.x * blockDim.x + threadIdx.x;
    int node = t >> 6;  // / D_OUT
    if (node < n) {
        float di = dinv[node];
        out[t] = h[t] * di * di;
    }
}

// 6) per-edge scatter: out[ta][d] += h[fa][d] * dinv[fa]*dinv[ta]
//    64 threads per edge -> wave-coalesced 128B reads of h[fa], f32 atomics
//    land in L2 (h and out both fit in the 192MB L2).
__global__ void scatter_kernel(const int* __restrict__ from_idx,
                               const int* __restrict__ to_idx,
                               const float* __restrict__ h,
                               const float* __restrict__ dinv,
                               float* __restrict__ out, int e) {
    int t = blockIdx.x * blockDim.x + threadIdx.x;
    int edge = t >> 6;      // / D_OUT
    int d    = t & 63;      // % D_OUT
    if (edge < e) {
        int fa = from_idx[edge];
        int ta = to_idx[edge];
        float c = dinv[fa] * dinv[ta];
        atomicAdd(&out[(size_t)ta * D_OUT + d], h[(size_t)fa * D_OUT + d] * c);
    }
}

// ---------------------------------------------------------------------------
extern "C" void kernel_launch(void* const* d_in, const int* in_sizes, int n_in,
                              void* d_out, int out_size, void* d_ws, size_t ws_size,
                              hipStream_t stream) {
    const float* x  = (const float*)d_in[0];   // (N, 128)
    const float* W  = (const float*)d_in[1];   // (64, 128)
    const float* b  = (const float*)d_in[2];   // (64,)
    const int*   ei = (const int*)d_in[3];     // (2, E) flat

    const int N = in_sizes[0] / D_IN;
    const int E = in_sizes[3] / 2;
    const int* from_idx = ei;
    const int* to_idx   = ei + E;

    float* out = (float*)d_out;

    // Workspace layout: [deg/dinv : N floats][h : N*64 floats], 256B aligned.
    float* deg = (float*)d_ws;
    size_t deg_bytes = ((size_t)N * sizeof(float) + 255) & ~(size_t)255;
    float* h = (float*)((char*)d_ws + deg_bytes);

    const int B256 = 256;

    deg_init_kernel<<<(N + B256 - 1) / B256, B256, 0, stream>>>(deg, N);
    deg_count_kernel<<<(E + B256 - 1) / B256, B256, 0, stream>>>(to_idx, deg, E);
    dinv_kernel<<<(N + B256 - 1) / B256, B256, 0, stream>>>(deg, N);

    // One block (4 waves) per 16-node tile; each wave covers 16 of 64 outputs.
    int m_tiles = (N + 15) / 16;
    gemm_wmma_kernel<<<m_tiles, 128, 0, stream>>>(x, W, b, h, N);

    long long self_threads = (long long)N * D_OUT;
    selfloop_kernel<<<(int)((self_threads + B256 - 1) / B256), B256, 0, stream>>>(
        h, deg, out, N);

    long long edge_threads = (long long)E * D_OUT;
    scatter_kernel<<<(int)((edge_threads + B256 - 1) / B256), B256, 0, stream>>>(
        from_idx, to_idx, h, deg, out, E);
}